// DogDetector_18236431139268
// MI455X (gfx1250) — compile-verified
//
#include <hip/hip_runtime.h>
#include <hip/hip_bf16.h>

// DogDetector post-process for MI455X (gfx1250, wave32).
// One 1024-thread workgroup (32 waves) per image; all intermediate state LDS-resident.
// bbox stream staged via CDNA5 async global->LDS DMA (ASYNCcnt), double buffered.
// Division-free filter + NMS (threshold comparisons restated as multiplies).

#define NANCH     6144
#define CHUNK     1024
#define NCHUNK    (NANCH / CHUNK)
#define CAP       1024          // max surviving candidates (expected ~600 for this input dist)
#define MAXDET    100
#define CONF_TH   0.5f
#define NMS_TH    0.5f

__global__ __launch_bounds__(1024, 1)
void dog_nms_kernel(const float* __restrict__ bbox,
                    const float* __restrict__ conf,
                    float* __restrict__ out, int batch) {
    __shared__ float4 s_stage[2][CHUNK];                 // 32 KB async staging (double buffer)
    __shared__ float  c_x1[CAP], c_y1[CAP], c_x2[CAP], c_y2[CAP], c_ar[CAP]; // 20 KB (+area)
    __shared__ unsigned long long s_key[CAP];            // 8 KB sort keys (score recoverable)
    __shared__ int s_wcnt[32];                           // per-wave valid counts / excl offsets
    __shared__ int s_base;                               // running compacted count
    __shared__ int s_next;                               // argmin for NMS head select
    __shared__ int s_kept[MAXDET];

    const int tid  = threadIdx.x;
    const int lane = tid & 31;          // wave32
    const int img  = blockIdx.x;

    const float* cimg = conf + (size_t)img * NANCH;
    const unsigned long long bbase =
        (unsigned long long)(size_t)(bbox + (size_t)img * NANCH * 4);

    // init sort keys to +inf padding (overwritten for compacted slots)
    s_key[tid] = ~0ull;

    // ---- prologue: async-DMA chunk 0 into staging buffer 0 ----------------
    {
        // generic LDS pointer low 32 bits == LDS byte offset (aperture in high bits)
        unsigned lds_addr = (unsigned)(size_t)&s_stage[0][tid];
        unsigned goff     = (unsigned)(tid * 16);
        asm volatile("global_load_async_to_lds_b128 %0, %1, %2"
                     :: "v"(lds_addr), "v"(goff), "s"(bbase) : "memory");
    }

    int base = 0;
    for (int c = 0; c < NCHUNK; ++c) {
        if (c + 1 < NCHUNK) {
            // issue next chunk, then drain down to 1 outstanding => chunk c complete
            unsigned lds_addr = (unsigned)(size_t)&s_stage[(c + 1) & 1][tid];
            unsigned goff     = (unsigned)(((c + 1) * CHUNK + tid) * 16);
            asm volatile("global_load_async_to_lds_b128 %0, %1, %2"
                         :: "v"(lds_addr), "v"(goff), "s"(bbase) : "memory");
            asm volatile("s_wait_asynccnt 1" ::: "memory");
        } else {
            asm volatile("s_wait_asynccnt 0" ::: "memory");
        }
        __syncthreads();                       // chunk c visible to all 32 waves

        float4 bx = s_stage[c & 1][tid];
        float  sc = cimg[c * CHUNK + tid];
        float x1 = fminf(fmaxf(bx.x, 0.f), 1.f);
        float y1 = fminf(fmaxf(bx.y, 0.f), 1.f);
        float x2 = fminf(fmaxf(bx.z, 0.f), 1.f);
        float y2 = fminf(fmaxf(bx.w, 0.f), 1.f);
        float w = x2 - x1, h = y2 - y1;
        float hd = h + 1e-6f;                  // > 0 whenever h > 0.01 holds below
        // division-free: (w/hd > 0.2 && w/hd < 5.0) <=> (w > 0.2*hd && w < 5.0*hd)
        bool flag = (sc > CONF_TH) && (w > 0.01f) && (h > 0.01f) &&
                    (w > 0.2f * hd) && (w < 5.0f * hd);

        // deterministic order-preserving compaction: wave32 ballot + cross-wave scan
        unsigned mask = (unsigned)__ballot(flag);
        int wpre = __popc(mask & ((1u << lane) - 1u));
        if (lane == 0) s_wcnt[tid >> 5] = __popc(mask);
        __syncthreads();
        if (tid < 32) {                        // wave-0 inclusive scan of 32 counts
            int v = s_wcnt[tid], x = v;
            #pragma unroll
            for (int d = 1; d < 32; d <<= 1) {
                int y = __shfl_up(x, d, 32);
                if (tid >= d) x += y;
            }
            s_wcnt[tid] = x - v;               // exclusive offsets
            if (tid == 31) s_base = base + x;  // running total
        }
        __syncthreads();
        if (flag) {
            int pos = base + s_wcnt[tid >> 5] + wpre;
            if (pos < CAP) {
                c_x1[pos] = x1; c_y1[pos] = y1;
                c_x2[pos] = x2; c_y2[pos] = y2;
                c_ar[pos] = w * h;             // precomputed area
                // key: ascending u64 sort == descending score, stable on index
                s_key[pos] = ((unsigned long long)(~__float_as_uint(sc)) << 32)
                             | (unsigned)pos;
            }
        }
        __syncthreads();                       // also fences staging buffer reuse
        base = s_base;
    }

    const int M = base < CAP ? base : CAP;
    __syncthreads();

    // ---- in-LDS bitonic sort, 1024 keys, 1024 threads ----------------------
    for (unsigned kk = 2; kk <= CAP; kk <<= 1) {
        for (unsigned j = kk >> 1; j > 0; j >>= 1) {
            unsigned i = (unsigned)tid, ixj = i ^ j;
            if (ixj > i) {
                unsigned long long a = s_key[i], b = s_key[ixj];
                bool up = ((i & kk) == 0);
                if ((a > b) == up) { s_key[i] = b; s_key[ixj] = a; }
            }
            __syncthreads();
        }
    }

    // ---- gather own candidate into registers (thread tid owns sorted pos tid) ----
    float rx1 = 0.f, ry1 = 0.f, rx2 = 0.f, ry2 = 0.f, rar = 0.f;
    bool  supp = (tid >= M);                   // register-resident suppression flag
    if (tid < M) {
        int j = (unsigned)s_key[tid];
        rx1 = c_x1[j]; ry1 = c_y1[j]; rx2 = c_x2[j]; ry2 = c_y2[j]; rar = c_ar[j];
    }

    // ---- greedy score-priority NMS: one round per kept head ----------------
    int keep_count = 0;
    for (;;) {
        if (tid == 0) s_next = 0x7fffffff;
        __syncthreads();
        if (!supp) atomicMin(&s_next, tid);
        __syncthreads();
        int h = s_next;
        if (h == 0x7fffffff || keep_count >= MAXDET) break;

        // head box: same-address LDS reads broadcast to all lanes
        int hj = (unsigned)s_key[h];
        float hx1 = c_x1[hj], hy1 = c_y1[hj], hx2 = c_x2[hj], hy2 = c_y2[hj];
        float harea = c_ar[hj];
        if (tid == 0) s_kept[keep_count] = h;

        if (!supp) {                           // all unsuppressed positions are > h (or == h)
            if (tid == h) {
                supp = true;                   // consumed as kept head
            } else {
                float iw = fmaxf(fminf(hx2, rx2) - fmaxf(hx1, rx1), 0.f);
                float ih = fmaxf(fminf(hy2, ry2) - fmaxf(hy1, ry1), 0.f);
                float inter = iw * ih;
                // division-free: inter/(U+eps) > thr  <=>  inter > thr*(U+eps), U+eps > 0
                if (inter > NMS_TH * (harea + rar - inter + 1e-6f)) supp = true;
            }
        }
        keep_count++;
        __syncthreads();
    }
    __syncthreads();

    // ---- emit top-100 (kept heads are already in descending-score order) ---
    float* ob = out + (size_t)img * MAXDET * 4;
    float* os = out + (size_t)batch * MAXDET * 4 + (size_t)img * MAXDET;
    float* ov = out + (size_t)batch * MAXDET * 5 + (size_t)img * MAXDET;
    if (tid < MAXDET) {
        if (tid < keep_count) {
            int p = s_kept[tid];
            unsigned long long k = s_key[p];
            int j = (unsigned)k;
            ob[tid * 4 + 0] = c_x1[j];
            ob[tid * 4 + 1] = c_y1[j];
            ob[tid * 4 + 2] = c_x2[j];
            ob[tid * 4 + 3] = c_y2[j];
            os[tid] = __uint_as_float(~(unsigned)(k >> 32));   // exact score bits from key
            ov[tid] = 1.0f;
        } else {
            ob[tid * 4 + 0] = 0.f; ob[tid * 4 + 1] = 0.f;
            ob[tid * 4 + 2] = 0.f; ob[tid * 4 + 3] = 0.f;
            os[tid] = 0.f;
            ov[tid] = 0.f;
        }
    }
}

extern "C" void kernel_launch(void* const* d_in, const int* in_sizes, int n_in,
                              void* d_out, int out_size, void* d_ws, size_t ws_size,
                              hipStream_t stream) {
    const float* bbox = (const float*)d_in[0];   // [B, N, 4] f32
    const float* conf = (const float*)d_in[1];   // [B, N]    f32
    float* out = (float*)d_out;                  // boxes[B,100,4] | scores[B,100] | valid[B,100]
    int batch = in_sizes[1] / NANCH;
    dog_nms_kernel<<<dim3(batch), dim3(1024), 0, stream>>>(bbox, conf, out, batch);
}